// DictionaryMatchOp_79310866088173
// MI455X (gfx1250) — compile-verified
//
#include <hip/hip_runtime.h>

// ---------------------------------------------------------------------------
// DictionaryMatchOp for MI455X (gfx1250, wave32, WMMA)
//
// sim[n,b] = (y_r^T in_r)^2 + (y_i^T in_r)^2 + (y_r^T in_i)^2 + (y_i^T in_i)^2
// idx[b]   = argmax_n sim[n,b]
// out      = [scale_r, scale_i, x_t1[idx], x_t2[idx]]  (4 x 8192)
//
// Compute-bound fp32 problem (84 GFLOP, ~15 MB resident -> lives in 192MB L2).
// Use V_WMMA_F32_16X16X4_F32 to keep fp32 accumulation (argmax stability).
// ---------------------------------------------------------------------------

#define M_TIME 64
#define N_DICT 20000
#define B_VOX  8192

#define NPART        25          // n-partitions (grid.y of kernel 1)
#define CHUNKS_PART  25          // 32-atom chunks per partition: 25*25*32 = 20000
#define WAVES_PER_WG 8
#define B_PER_WG     (WAVES_PER_WG * 16)   // 128 voxels per workgroup

typedef __attribute__((ext_vector_type(2))) float v2f;
typedef __attribute__((ext_vector_type(8))) float v8f;

// D = A(16x4 f32) * B(4x16 f32) + C(16x16 f32)
// probe pattern for 16x16x{4,32}: (neg_a, A, neg_b, B, c_mod, C, reuse_a, reuse_b)
#define WMMA_F32(A, B, C) \
    __builtin_amdgcn_wmma_f32_16x16x4_f32(false, (A), false, (B), (short)0, (C), false, false)

// ---------------------------------------------------------------------------
// Kernel 1: WMMA similarity + per-(voxel, partition) partial argmax
// grid = (B_VOX/B_PER_WG, NPART), block = 256 (8 waves)
// Each wave: 16 voxels, loops over CHUNKS_PART chunks of 32 dictionary atoms.
// ---------------------------------------------------------------------------
__global__ __launch_bounds__(256)
void dict_sim_argmax_kernel(const float* __restrict__ in_r,
                            const float* __restrict__ in_i,
                            const float* __restrict__ y_r,
                            const float* __restrict__ y_i,
                            float* __restrict__ part_max,
                            int*   __restrict__ part_idx)
{
    const int lane = threadIdx.x & 31;          // wave32
    const int wave = threadIdx.x >> 5;          // 0..7
    const int half = lane >> 4;                 // 0|1
    const int l16  = lane & 15;

    const int b0   = blockIdx.x * B_PER_WG + wave * 16;  // this wave's voxel base
    const int part = blockIdx.y;
    const int n_start = part * CHUNKS_PART * 32;

    float bestVal = -1.0f;                      // sim >= 0
    int   bestIdx = 0;

    for (int c = 0; c < CHUNKS_PART; ++c) {
        const int n0 = n_start + c * 32;

        v8f acc[2][4];                          // [n-tile][rr, ir, ri, ii]
        #pragma unroll
        for (int t = 0; t < 2; ++t)
            #pragma unroll
            for (int p = 0; p < 4; ++p)
                acc[t][p] = (v8f)0.0f;

        #pragma unroll
        for (int k = 0; k < M_TIME; k += 4) {
            const int ko = k + 2 * half;        // K rows for vgpr0/vgpr1 of A and B

            // B operands: inputs, N = voxel, K = time (coalesced 16-lane rows)
            v2f br, bi;
            br.x = in_r[(ko + 0) * B_VOX + b0 + l16];
            br.y = in_r[(ko + 1) * B_VOX + b0 + l16];
            bi.x = in_i[(ko + 0) * B_VOX + b0 + l16];
            bi.y = in_i[(ko + 1) * B_VOX + b0 + l16];

            #pragma unroll
            for (int t = 0; t < 2; ++t) {
                const int nb = n0 + t * 16;
                // A operands: dictionary, M = atom, K = time
                v2f ar, ai;
                ar.x = y_r[(ko + 0) * N_DICT + nb + l16];
                ar.y = y_r[(ko + 1) * N_DICT + nb + l16];
                ai.x = y_i[(ko + 0) * N_DICT + nb + l16];
                ai.y = y_i[(ko + 1) * N_DICT + nb + l16];

                acc[t][0] = WMMA_F32(ar, br, acc[t][0]);   // y_r . in_r
                acc[t][1] = WMMA_F32(ai, br, acc[t][1]);   // y_i . in_r
                acc[t][2] = WMMA_F32(ar, bi, acc[t][2]);   // y_r . in_i
                acc[t][3] = WMMA_F32(ai, bi, acc[t][3]);   // y_i . in_i
            }
        }

        // sim + per-lane argmax. C/D layout: vgpr r holds row M = r + 8*half,
        // column N = l16. n increases with r, so strict '>' keeps lowest n.
        #pragma unroll
        for (int t = 0; t < 2; ++t) {
            #pragma unroll
            for (int r = 0; r < 8; ++r) {
                const float s = acc[t][0][r] * acc[t][0][r]
                              + acc[t][1][r] * acc[t][1][r]
                              + acc[t][2][r] * acc[t][2][r]
                              + acc[t][3][r] * acc[t][3][r];
                const int n = n0 + t * 16 + r + 8 * half;
                if (s > bestVal) { bestVal = s; bestIdx = n; }
            }
        }
    }

    // Fold the two half-wave row sets (same voxel, disjoint atoms).
    {
        const float oVal = __shfl_xor(bestVal, 16, 32);
        const int   oIdx = __shfl_xor(bestIdx, 16, 32);
        if (oVal > bestVal || (oVal == bestVal && oIdx < bestIdx)) {
            bestVal = oVal; bestIdx = oIdx;
        }
    }

    if (lane < 16) {
        const int b = b0 + l16;
        part_max[part * B_VOX + b] = bestVal;
        part_idx[part * B_VOX + b] = bestIdx;
    }
}

// ---------------------------------------------------------------------------
// Kernel 2: reduce partials -> final idx, then gather dot products + outputs
// ---------------------------------------------------------------------------
__global__ __launch_bounds__(256)
void dict_finalize_kernel(const float* __restrict__ in_r,
                          const float* __restrict__ in_i,
                          const float* __restrict__ y_r,
                          const float* __restrict__ y_i,
                          const float* __restrict__ inv_norm,
                          const float* __restrict__ x_t1,
                          const float* __restrict__ x_t2,
                          const float* __restrict__ part_max,
                          const int*   __restrict__ part_idx,
                          float* __restrict__ out)
{
    const int b = blockIdx.x * blockDim.x + threadIdx.x;
    if (b >= B_VOX) return;

    // Partitions are ordered by n; strict '>' keeps the first (lowest-n) max.
    float bestVal = -1.0f;
    int   bestIdx = 0;
    for (int p = 0; p < NPART; ++p) {
        const float v = part_max[p * B_VOX + b];
        if (v > bestVal) { bestVal = v; bestIdx = part_idx[p * B_VOX + b]; }
    }

    float sr = 0.0f, si = 0.0f;
    #pragma unroll 8
    for (int m = 0; m < M_TIME; ++m) {
        const float yr = y_r[m * N_DICT + bestIdx];
        const float yi = y_i[m * N_DICT + bestIdx];
        const float ir = in_r[m * B_VOX + b];
        const float ii = in_i[m * B_VOX + b];
        sr += yr * ir + yi * ii;
        si += yr * ii - yi * ir;
    }

    const float inv = inv_norm[bestIdx];
    out[0 * B_VOX + b] = sr * inv;
    out[1 * B_VOX + b] = si * inv;
    out[2 * B_VOX + b] = x_t1[bestIdx];
    out[3 * B_VOX + b] = x_t2[bestIdx];
}

// ---------------------------------------------------------------------------
extern "C" void kernel_launch(void* const* d_in, const int* in_sizes, int n_in,
                              void* d_out, int out_size, void* d_ws, size_t ws_size,
                              hipStream_t stream)
{
    const float* in_r     = (const float*)d_in[0];
    const float* in_i     = (const float*)d_in[1];
    const float* y_r      = (const float*)d_in[2];
    const float* y_i      = (const float*)d_in[3];
    const float* inv_norm = (const float*)d_in[4];
    const float* x_t1     = (const float*)d_in[5];
    const float* x_t2     = (const float*)d_in[6];
    float* out = (float*)d_out;

    float* part_max = (float*)d_ws;                              // NPART * B_VOX floats
    int*   part_idx = (int*)((char*)d_ws + (size_t)NPART * B_VOX * sizeof(float));

    dim3 grid1(B_VOX / B_PER_WG, NPART);                         // (64, 25)
    dict_sim_argmax_kernel<<<grid1, 256, 0, stream>>>(in_r, in_i, y_r, y_i,
                                                      part_max, part_idx);

    dict_finalize_kernel<<<B_VOX / 256, 256, 0, stream>>>(in_r, in_i, y_r, y_i,
                                                          inv_norm, x_t1, x_t2,
                                                          part_max, part_idx, out);
}